// CAT_FullAttention_18915035972061
// MI455X (gfx1250) — compile-verified
//
#include <hip/hip_runtime.h>
#include <hip/hip_bf16.h>
#include <math.h>

// Problem constants (hardcoded by the reference's reshapes)
#define BB 256
#define LL 96
#define RR 70
#define SSn 96
#define EE 10
#define PP 7

#define LT 4                 // l's per tile
#define NLT (LL / LT)        // 24 l-tiles
#define LGROUPS 3            // blocks per batch
#define ITERS (NLT / LGROUPS)// 8 l-tiles per block: K/V staged once, reused 8x
#define ROWS (LT * PP)       // 28 valid score rows (l,p)
#define ROWPAD 32            // padded to 2 wmma row-tiles
#define COLS (SSn * PP)      // 672 score cols (s,r) -- exactly 42 tiles of 16
#define KPAD 12              // E=10 padded to 3 k-steps of 4
#define KS_STRIDE 680        // 672 + 8: second half-wave lands on banks 16..31
#define SS_STRIDE 673        // 672 + 1: de-conflict row-strided accesses

typedef __attribute__((ext_vector_type(2))) float v2f;
typedef __attribute__((ext_vector_type(8))) float v8f;

#if defined(__has_builtin)
#if __has_builtin(__builtin_amdgcn_global_load_async_to_lds_b32) && \
    __has_builtin(__builtin_amdgcn_global_load_async_to_lds_b128)
#define USE_ASYNC_LDS 1
#endif
#endif

// Typed pointer casts matching the builtin signatures revealed by the compiler:
//  b32 : (addrspace(1) int*, addrspace(3) int*, imm, imm)
//  b128: (addrspace(1) int __vector_size(16)*, addrspace(3) ..., imm, imm)
typedef int v4i_vs __attribute__((vector_size(16)));
#define GLB_I32(p) ((__attribute__((address_space(1))) int*)(p))
#define LDS_I32(p) ((__attribute__((address_space(3))) int*)(p))
#define GLB_V4I(p) ((__attribute__((address_space(1))) v4i_vs*)(p))
#define LDS_V4I(p) ((__attribute__((address_space(3))) v4i_vs*)(p))

__device__ __forceinline__ void wait_async_lds() {
#if defined(USE_ASYNC_LDS)
#if __has_builtin(__builtin_amdgcn_s_wait_asynccnt)
  __builtin_amdgcn_s_wait_asynccnt(0);
#else
  asm volatile("s_wait_asynccnt 0" ::: "memory");
#endif
#endif
}

__global__ __launch_bounds__(256) void cat_attn_kernel(
    const float* __restrict__ q, const float* __restrict__ k,
    const float* __restrict__ v, float* __restrict__ out) {
  __shared__ float Qs[ROWPAD * KPAD];                    //   1.5 KB  (l,p) x e
  __shared__ __align__(16) float Ks[KPAD * KS_STRIDE];   //  31.9 KB  e x (s,r)
  __shared__ __align__(16) float Vs[SSn * RR];           //  26.3 KB  raw v[b]
  __shared__ float Ssc[ROWPAD * SS_STRIDE];              //  84.1 KB  scores (l,p) x (s,r)
  __shared__ float Dacc[2 * 256];                        //   2.0 KB  split-K accumulators
  __shared__ float invS[ROWS];                           //  softmax 1/sum per (l,r)

  const int b = blockIdx.y;
  const int tid = threadIdx.x;
  const int wave = tid >> 5;
  const int lane = tid & 31;

  const float* kb = k + (size_t)b * LL * RR;
  const float* vb = v + (size_t)b * LL * RR;
  const float* qbase = q + (size_t)b * LL * RR;

  // ---------------- Stage K (transposed e x (s,r)) and V once per block ----------------
#if defined(USE_ASYNC_LDS)
  for (int i = tid; i < EE * COLS; i += 256) {
    int e = i / COLS, c = i % COLS;
    int s = c / PP, r = c % PP;
    __builtin_amdgcn_global_load_async_to_lds_b32(
        GLB_I32(kb + s * RR + e * PP + r), LDS_I32(&Ks[e * KS_STRIDE + c]), 0, 0);
  }
  for (int j = tid; j < (SSn * RR) / 4; j += 256)
    __builtin_amdgcn_global_load_async_to_lds_b128(
        GLB_V4I(vb + 4 * j), LDS_V4I(&Vs[4 * j]), 0, 0);
#else
  for (int i = tid; i < EE * COLS; i += 256) {
    int e = i / COLS, c = i % COLS;
    int s = c / PP, r = c % PP;
    Ks[e * KS_STRIDE + c] = kb[s * RR + e * PP + r];
  }
  for (int i = tid; i < SSn * RR; i += 256) Vs[i] = vb[i];
#endif
  for (int i = tid; i < 2 * COLS; i += 256)  // zero-pad e = 10, 11 rows
    Ks[(EE + i / COLS) * KS_STRIDE + (i % COLS)] = 0.0f;
  wait_async_lds();
  __syncthreads();

  for (int it = 0; it < ITERS; ++it) {
    const int lt = blockIdx.x * ITERS + it;
    const int l0 = lt * LT;
    const float* qb = qbase + (size_t)l0 * RR;

    // ---------------- Stage Q tile, zero accumulators, prefetch next Q ----------------
    for (int i = tid; i < ROWPAD * KPAD; i += 256) {
      int row = i / KPAD, e = i % KPAD;
      int l = row / PP, p = row % PP;
      float val = 0.0f;
      if (row < ROWS && e < EE) val = qb[l * RR + e * PP + p];
      Qs[row * KPAD + e] = val;
    }
    for (int i = tid; i < 2 * 256; i += 256) { Dacc[i] = 0.0f; Dacc[i + 256] = 0.0f; }
    if (it + 1 < ITERS && tid < 9)  // next Q tile: 1120 B -> global_prefetch_b8
      __builtin_prefetch(qb + LT * RR + tid * 32, 0, 0);
    __syncthreads();

    // ---------------- GEMM1: scores = Q x K^T  (M=32, N=672, K=12) ----------------
    {
      const int khalf = (lane >> 4) * 2;  // lanes 16-31 hold K+2,K+3
      for (int t = wave; t < 2 * (COLS / 16); t += 8) {
        const int ct = t >> 1;
        const int rt = t & 1;
        const int m = rt * 16 + (lane & 15);
        const int n = ct * 16 + (lane & 15);
        v8f c = {};
#pragma unroll
        for (int k0 = 0; k0 < KPAD; k0 += 4) {
          v2f a, bmat;
          a.x = Qs[m * KPAD + k0 + khalf];
          a.y = Qs[m * KPAD + k0 + khalf + 1];
          bmat.x = Ks[(k0 + khalf) * KS_STRIDE + n];
          bmat.y = Ks[(k0 + khalf + 1) * KS_STRIDE + n];
          c = __builtin_amdgcn_wmma_f32_16x16x4_f32(false, a, false, bmat,
                                                    (short)0, c, false, false);
        }
        const int srow = rt * 16 + ((lane >> 4) ? 8 : 0);
        const int scol = ct * 16 + (lane & 15);
#pragma unroll
        for (int j = 0; j < 8; ++j)
          Ssc[(srow + j) * SS_STRIDE + scol] = c[j];
      }
    }
    __syncthreads();

    // ---- Softmax stats over (s,p) per (l,r): store exp in place, fold 1/sum into epilogue ----
    for (int g = wave; g < ROWS; g += 8) {
      const int l = g / PP, r = g % PP;
      const int rowb = (l * PP) * SS_STRIDE + r;
      float mx = -INFINITY;
#pragma unroll
      for (int p = 0; p < PP; ++p)
        for (int s = lane; s < SSn; s += 32)
          mx = fmaxf(mx, Ssc[rowb + p * SS_STRIDE + s * PP]);
#pragma unroll
      for (int off = 16; off > 0; off >>= 1) mx = fmaxf(mx, __shfl_xor(mx, off, 32));
      float sum = 0.0f;
#pragma unroll
      for (int p = 0; p < PP; ++p)
        for (int s = lane; s < SSn; s += 32) {
          const int idx = rowb + p * SS_STRIDE + s * PP;
          const float e = __expf(Ssc[idx] - mx);
          Ssc[idx] = e;
          sum += e;
        }
#pragma unroll
      for (int off = 16; off > 0; off >>= 1) sum += __shfl_xor(sum, off, 32);
      if (lane == 0) invS[g] = 1.0f / sum;
    }
    __syncthreads();

    // ---------------- GEMM2: out = P x V  (M=(l,r)=28, N=e=10, K=(s,p)=672) ----------------
    {
      const int tile = wave >> 2;       // output row-tile 0..1
      const int kq = wave & 3;          // split-K quarter
      const int mm = tile * 16 + (lane & 15);
      const int lidx = mm / PP, ridx = mm % PP;
      const bool mvalid = mm < ROWS;
      const int nn = lane & 15;         // e
      const bool nvalid = nn < EE;
      const int khalf = (lane >> 4) * 2;
      const int baseA = (lidx * PP) * SS_STRIDE + ridx;
      v8f c = {};
      int kk = kq * 168 + khalf;        // k = s*7 + p, advances by 4 each step
      int s0 = kk / PP, p0 = kk % PP;
      int s1 = (kk + 1) / PP, p1 = (kk + 1) % PP;
      for (int ks = 0; ks < 42; ++ks) {
        v2f a, bmat;
        a.x = mvalid ? Ssc[baseA + p0 * SS_STRIDE + s0 * PP] : 0.0f;
        a.y = mvalid ? Ssc[baseA + p1 * SS_STRIDE + s1 * PP] : 0.0f;
        bmat.x = nvalid ? Vs[s0 * RR + nn * PP + p0] : 0.0f;
        bmat.y = nvalid ? Vs[s1 * RR + nn * PP + p1] : 0.0f;
        c = __builtin_amdgcn_wmma_f32_16x16x4_f32(false, a, false, bmat,
                                                  (short)0, c, false, false);
        p0 += 4; if (p0 >= PP) { p0 -= PP; ++s0; }   // k += 4, no division
        p1 += 4; if (p1 >= PP) { p1 -= PP; ++s1; }
      }
      const int r0 = (lane >> 4) ? 8 : 0;
      const int col = lane & 15;
#pragma unroll
      for (int j = 0; j < 8; ++j)
        atomicAdd(&Dacc[tile * 256 + (r0 + j) * 16 + col], c[j]);  // ds_add_f32
    }
    __syncthreads();

    // ---------------- Epilogue: scale by 1/sum and scatter ----------------
    for (int i = tid; i < ROWS * EE; i += 256) {
      const int m = i / EE;             // (l,r) row
      const int e = i % EE;
      const int l = m / PP, r = m % PP;
      out[(size_t)b * LL * RR + (size_t)(l0 + l) * RR + e * PP + r] =
          Dacc[(m >> 4) * 256 + (m & 15) * 16 + e] * invS[m];
    }
    __syncthreads();  // protect Qs/Dacc/invS before next iteration rewrites them
  }
}

extern "C" void kernel_launch(void* const* d_in, const int* in_sizes, int n_in,
                              void* d_out, int out_size, void* d_ws, size_t ws_size,
                              hipStream_t stream) {
  const float* q = (const float*)d_in[0];
  const float* k = (const float*)d_in[1];
  const float* v = (const float*)d_in[2];
  // d_in[3] = attn_mask, unused (mask_flag=False in reference)
  float* out = (float*)d_out;
  dim3 grid(LGROUPS, BB);
  cat_attn_kernel<<<grid, 256, 0, stream>>>(q, k, v, out);
}